// TripletLoss_20469814132850
// MI455X (gfx1250) — compile-verified
//
#include <hip/hip_runtime.h>
#include <hip/hip_bf16.h>
#include <math.h>

#define NN 8192
#define DD 1024
#define DD2 512
#define MRG 0.3f
#define BIGV 9999999.0f
#define NEG_INF -3.0e38f
#define POS_INF 3.0e38f

typedef __attribute__((ext_vector_type(2))) float v2f;
typedef __attribute__((ext_vector_type(8))) float v8f;

// ---------------- row squared-norms ----------------
__global__ __launch_bounds__(128) void rownorm2_kernel(const float* __restrict__ x,
                                                       int d, float* __restrict__ out) {
  __shared__ float red[128];
  int row = blockIdx.x;
  const float* xr = x + (size_t)row * d;
  float s = 0.f;
  for (int k = threadIdx.x; k < d; k += 128) { float v = xr[k]; s += v * v; }
  red[threadIdx.x] = s;
  __syncthreads();
  for (int off = 64; off > 0; off >>= 1) {
    if ((int)threadIdx.x < off) red[threadIdx.x] += red[threadIdx.x + off];
    __syncthreads();
  }
  if (threadIdx.x == 0) out[row] = red[0];
}

// ------------- fused dist-GEMM + hard mining (WMMA f32) -------------
// Block = 256 threads (8 waves), owns 32 rows (two 16-row strips in 128 KB LDS).
// Each wave streams one 16-col B tile and feeds it to BOTH strips' WMMAs.
// A and B fragments are double-buffered across k-iterations; a compiler memory
// barrier per tile stops LICM from hoisting LDS fragments (which caused
// catastrophic scratch spills).
__global__ __launch_bounds__(256) void dist_topk_kernel(
    const float* __restrict__ emb, const int* __restrict__ label,
    const float* __restrict__ sq,
    float* __restrict__ p1v, int* __restrict__ i1g,
    float* __restrict__ p2v, int* __restrict__ i2g,
    float* __restrict__ ngv) {
  // [strip][k2][r] layout; per strip 512*16 float2 = 64KB, bank-conflict-free
  __shared__ float2 ldsA[2][(DD / 2) * 16];

  const int rowBase = blockIdx.x * 32;

  for (int idx = threadIdx.x; idx < 2 * (DD / 2) * 16; idx += 256) {
    int t = idx >> 13;  // strip
    int e = idx & 8191;
    int r = e & 15;
    int k2 = e >> 4;
    ldsA[t][e] = *(const float2*)(emb + (size_t)(rowBase + t * 16 + r) * DD + 2 * k2);
  }
  __syncthreads();

  const int lane = threadIdx.x & 31;
  const int wave = threadIdx.x >> 5;
  const int half = lane >> 4;
  const int ln = lane & 15;

  float sqr[2][8];
  int lbr[2][8];
#pragma unroll
  for (int t = 0; t < 2; t++)
#pragma unroll
    for (int v = 0; v < 8; v++) {
      int m = rowBase + t * 16 + v + half * 8;  // C/D layout: M = vgpr + (lane/16)*8
      sqr[t][v] = sq[m];
      lbr[t][v] = label[m];
    }

  float p1[2][8], p2[2][8], nv[2][8];
  int i1[2][8], i2[2][8];
#pragma unroll
  for (int t = 0; t < 2; t++)
#pragma unroll
    for (int v = 0; v < 8; v++) {
      p1[t][v] = NEG_INF; p2[t][v] = NEG_INF; nv[t][v] = POS_INF;
      i1[t][v] = 0; i2[t][v] = 0;
    }

  for (int ct = wave; ct < NN / 16; ct += 8) {
    // stop LICM from hoisting the (ct-invariant) LDS fragment loads out of
    // this loop -- that blows up VGPR pressure and spills to scratch.
    asm volatile("" ::: "memory");

    const int col = ct * 16 + ln;
    const float* bptr = emb + (size_t)col * DD + half * 2;
    if (col + 128 < NN)  // prefetch this lane's row for the wave's next tile
      __builtin_prefetch((const void*)(emb + (size_t)(col + 128) * DD), 0, 1);

    v8f acc0 = {0.f, 0.f, 0.f, 0.f, 0.f, 0.f, 0.f, 0.f};
    v8f acc1 = {0.f, 0.f, 0.f, 0.f, 0.f, 0.f, 0.f, 0.f};

    // prime double buffers (K = half*2 .. +1 at k0 = 0)
    v2f b = *(const v2f*)(bptr);
    float2 a0c = ldsA[0][half * 16 + ln];
    float2 a1c = ldsA[1][half * 16 + ln];

#pragma unroll 4
    for (int k0 = 0; k0 < DD - 4; k0 += 4) {
      v2f bn = *(const v2f*)(bptr + k0 + 4);  // next iteration's operands
      float2 a0n = ldsA[0][(((k0 + 4) >> 1) + half) * 16 + ln];
      float2 a1n = ldsA[1][(((k0 + 4) >> 1) + half) * 16 + ln];
      v2f a0 = {a0c.x, a0c.y};
      v2f a1 = {a1c.x, a1c.y};
      acc0 = __builtin_amdgcn_wmma_f32_16x16x4_f32(false, a0, false, b,
                                                   (short)0, acc0, false, false);
      acc1 = __builtin_amdgcn_wmma_f32_16x16x4_f32(false, a1, false, b,
                                                   (short)0, acc1, false, false);
      a0c = a0n; a1c = a1n; b = bn;
    }
    {  // tail: k0 = DD-4, operands already in registers
      v2f a0 = {a0c.x, a0c.y};
      v2f a1 = {a1c.x, a1c.y};
      acc0 = __builtin_amdgcn_wmma_f32_16x16x4_f32(false, a0, false, b,
                                                   (short)0, acc0, false, false);
      acc1 = __builtin_amdgcn_wmma_f32_16x16x4_f32(false, a1, false, b,
                                                   (short)0, acc1, false, false);
    }

    const float sqc = sq[col];
    const int lbc = label[col];
#pragma unroll
    for (int t = 0; t < 2; t++) {
#pragma unroll
      for (int v = 0; v < 8; v++) {
        float accv = (t == 0) ? acc0[v] : acc1[v];
        float d2 = sqr[t][v] + sqc - 2.0f * accv;
        float dd = sqrtf(fmaxf(d2, 1e-12f));
        bool same = (lbr[t][v] == lbc);
        float pv = same ? dd : dd - BIGV;    // dist - BIG*(1-sim)
        float nval = same ? dd + BIGV : dd;  // dist + BIG*sim
        if (pv > p1[t][v]) { p2[t][v] = p1[t][v]; i2[t][v] = i1[t][v];
                             p1[t][v] = pv; i1[t][v] = col; }
        else if (pv > p2[t][v]) { p2[t][v] = pv; i2[t][v] = col; }
        nv[t][v] = fminf(nv[t][v], nval);
      }
    }
  }

  // merge across the 16 lanes of each half (xor butterfly, top-2 aware)
#pragma unroll
  for (int mk = 1; mk <= 8; mk <<= 1) {
#pragma unroll
    for (int t = 0; t < 2; t++) {
#pragma unroll
      for (int v = 0; v < 8; v++) {
        float o1 = __shfl_xor(p1[t][v], mk, 32);
        int oi1 = __shfl_xor(i1[t][v], mk, 32);
        float o2 = __shfl_xor(p2[t][v], mk, 32);
        int oi2 = __shfl_xor(i2[t][v], mk, 32);
        float on = __shfl_xor(nv[t][v], mk, 32);
        if (o1 > p1[t][v]) { p2[t][v] = p1[t][v]; i2[t][v] = i1[t][v];
                             p1[t][v] = o1; i1[t][v] = oi1; }
        else if (o1 > p2[t][v]) { p2[t][v] = o1; i2[t][v] = oi1; }
        if (o2 > p2[t][v]) { p2[t][v] = o2; i2[t][v] = oi2; }  // o2 <= o1
        nv[t][v] = fminf(nv[t][v], on);
      }
    }
  }

  __syncthreads();  // everyone done reading ldsA -> reuse it as merge buffer
  float* mg = (float*)ldsA;
  int* mgi = (int*)ldsA;
  if (ln == 0) {  // lanes 0 and 16 of each wave
#pragma unroll
    for (int t = 0; t < 2; t++) {
#pragma unroll
      for (int v = 0; v < 8; v++) {
        int r = t * 16 + v + half * 8;  // row within block (0..31)
        int base = (r * 8 + wave) * 6;
        mg[base + 0] = p1[t][v];
        mg[base + 1] = p2[t][v];
        mg[base + 2] = nv[t][v];
        mgi[base + 3] = i1[t][v];
        mgi[base + 4] = i2[t][v];
      }
    }
  }
  __syncthreads();
  if (threadIdx.x < 32) {
    int r = threadIdx.x;
    float P1 = NEG_INF, P2 = NEG_INF, NV = POS_INF;
    int I1 = 0, I2 = 0;
    for (int w = 0; w < 8; w++) {
      int base = (r * 8 + w) * 6;
      float c1 = mg[base + 0], c2 = mg[base + 1], cn = mg[base + 2];
      int ci1 = mgi[base + 3], ci2 = mgi[base + 4];
      if (c1 > P1) { P2 = P1; I2 = I1; P1 = c1; I1 = ci1; }
      else if (c1 > P2) { P2 = c1; I2 = ci1; }
      if (c2 > P2) { P2 = c2; I2 = ci2; }
      NV = fminf(NV, cn);
    }
    int row = rowBase + r;
    p1v[row] = P1; p2v[row] = P2; ngv[row] = NV;
    i1g[row] = I1; i2g[row] = I2;
  }
}

// ------------- cos lookups at the two hardest positives only -------------
__global__ __launch_bounds__(32) void alpha_kernel(
    const float* __restrict__ clot, const float* __restrict__ nrm2,
    const int* __restrict__ i1g, const int* __restrict__ i2g,
    float* __restrict__ a1, float* __restrict__ a2) {
  int row = blockIdx.x;
  int lane = threadIdx.x;
  int j1 = i1g[row], j2 = i2g[row];
  const float4* xi = (const float4*)(clot + (size_t)row * DD2);
  const float4* x1 = (const float4*)(clot + (size_t)j1 * DD2);
  const float4* x2 = (const float4*)(clot + (size_t)j2 * DD2);
  float s1 = 0.f, s2 = 0.f;
  for (int q = lane; q < DD2 / 4; q += 32) {
    float4 a = xi[q], b = x1[q], c = x2[q];
    s1 += a.x * b.x + a.y * b.y + a.z * b.z + a.w * b.w;
    s2 += a.x * c.x + a.y * c.y + a.z * c.z + a.w * c.w;
  }
#pragma unroll
  for (int off = 16; off > 0; off >>= 1) {
    s1 += __shfl_xor(s1, off, 32);
    s2 += __shfl_xor(s2, off, 32);
  }
  if (lane == 0) {
    float ni = sqrtf(nrm2[row]);
    a1[row] = s1 / (ni * sqrtf(nrm2[j1]));
    a2[row] = s2 / (ni * sqrtf(nrm2[j2]));
  }
}

// ------------- final loss / precision reduction -------------
__global__ __launch_bounds__(256) void final_kernel(
    const float* __restrict__ p1v, const float* __restrict__ p2v,
    const float* __restrict__ ngv, const float* __restrict__ a1,
    const float* __restrict__ a2, float* __restrict__ out) {
  __shared__ float red[256];
  float s11 = 0.f, s13 = 0.f, sp = 0.f;
  for (int i = threadIdx.x; i < NN; i += 256) {
    float ap1 = p1v[i], ap2 = p2v[i], an = ngv[i];
    float A1 = a1[i], A2 = a2[i];
    float y = (A1 < A2) ? -1.f : 1.f;
    float ym = (A1 == A2) ? 0.f : 1.f;
    float x1 = ap2 * ym;
    float x2 = ap1 * ym + MRG * (A1 - A2 - y);
    s11 += fmaxf(0.f, -y * (x1 - x2) + MRG);
    float ap1m = ap1 + MRG * (A1 - 1.f);
    s13 += fmaxf(0.f, -(an - ap1m) + MRG);
    sp += (an > ap1m) ? 1.f : 0.f;
  }
  red[threadIdx.x] = s11;
  __syncthreads();
  for (int o = 128; o > 0; o >>= 1) {
    if ((int)threadIdx.x < o) red[threadIdx.x] += red[threadIdx.x + o];
    __syncthreads();
  }
  float t11 = red[0];
  __syncthreads();
  red[threadIdx.x] = s13;
  __syncthreads();
  for (int o = 128; o > 0; o >>= 1) {
    if ((int)threadIdx.x < o) red[threadIdx.x] += red[threadIdx.x + o];
    __syncthreads();
  }
  float t13 = red[0];
  __syncthreads();
  red[threadIdx.x] = sp;
  __syncthreads();
  for (int o = 128; o > 0; o >>= 1) {
    if ((int)threadIdx.x < o) red[threadIdx.x] += red[threadIdx.x + o];
    __syncthreads();
  }
  float tp = red[0];
  if (threadIdx.x == 0) {
    out[0] = 0.1f * (t11 / (float)NN) + (t13 / (float)NN);
    out[1] = tp / (float)NN;
  }
}

extern "C" void kernel_launch(void* const* d_in, const int* in_sizes, int n_in,
                              void* d_out, int out_size, void* d_ws, size_t ws_size,
                              hipStream_t stream) {
  const float* emb = (const float*)d_in[0];
  const int* label = (const int*)d_in[1];
  const float* clot = (const float*)d_in[2];
  float* out = (float*)d_out;

  float* sq = (float*)d_ws;        // NN
  float* nrm2 = sq + NN;           // NN
  float* p1v = nrm2 + NN;          // NN
  float* p2v = p1v + NN;           // NN
  float* ngv = p2v + NN;           // NN
  int* i1g = (int*)(ngv + NN);     // NN
  int* i2g = i1g + NN;             // NN
  float* a1 = (float*)(i2g + NN);  // NN
  float* a2 = a1 + NN;             // NN

  rownorm2_kernel<<<NN, 128, 0, stream>>>(emb, DD, sq);
  rownorm2_kernel<<<NN, 128, 0, stream>>>(clot, DD2, nrm2);
  dist_topk_kernel<<<NN / 32, 256, 0, stream>>>(emb, label, sq, p1v, i1g, p2v, i2g, ngv);
  alpha_kernel<<<NN, 32, 0, stream>>>(clot, nrm2, i1g, i2g, a1, a2);
  final_kernel<<<1, 256, 0, stream>>>(p1v, p2v, ngv, a1, a2, out);
}